// adknnLoss_74534862455478
// MI455X (gfx1250) — compile-verified
//
#include <hip/hip_runtime.h>

#define NROWS 8192
#define MFEAT 128
#define JSPLIT 8
#define JTILES_PER_SPLIT (NROWS / 16 / JSPLIT)   // 64

typedef __attribute__((ext_vector_type(2))) float v2f;
typedef __attribute__((ext_vector_type(8))) float v8f;

// --------------------------------------------------------------------------
// Kernel 1: Bpre = X @ Wnet   (f32 WMMA 16x16x4, one 16x16 tile per wave)
// A operand layout (16x4 f32): lane l holds row M=l&15; regs {0,1} hold
// K = 2*(l>>4) + {0,1}  -> contiguous float2 load from X.
// B operand (4x16): lane l holds col N=l&15; regs hold K = 2*(l>>4)+{0,1}
// -> two strided loads from Wnet (row-major [K][C]).
// --------------------------------------------------------------------------
__global__ __launch_bounds__(128) void k_gemm_xw(const float* __restrict__ X,
                                                 const float* __restrict__ W,
                                                 float* __restrict__ Bpre) {
  const int lane  = threadIdx.x & 31;
  const int wave  = threadIdx.x >> 5;
  const int tile  = blockIdx.x * 4 + wave;   // 4096 tiles: 512 i-tiles x 8 c-tiles
  const int iBase = (tile >> 3) * 16;
  const int cBase = (tile & 7) * 16;
  const int half  = lane >> 4;
  const int lm    = lane & 15;

  v8f c = {};
#pragma unroll
  for (int ks = 0; ks < 32; ++ks) {
    const int k0 = ks * 4 + 2 * half;
    v2f a = *(const v2f*)(X + (iBase + lm) * MFEAT + k0);
    v2f b;
    b.x = W[(k0 + 0) * MFEAT + cBase + lm];
    b.y = W[(k0 + 1) * MFEAT + cBase + lm];
    c = __builtin_amdgcn_wmma_f32_16x16x4_f32(false, a, false, b, (short)0, c,
                                              false, false);
  }
#pragma unroll
  for (int v = 0; v < 8; ++v)
    Bpre[(iBase + v + 8 * half) * MFEAT + cBase + lm] = c[v];
}

// --------------------------------------------------------------------------
// Kernel 2: A = L*X ; B = (Bpre + bnet)*X (in place over Bpre);
//           NA[i]=||A_i||^2, NB[i]=||B_i||^2. One wave per row.
// --------------------------------------------------------------------------
__global__ __launch_bounds__(256) void k_prep(const float* __restrict__ L,
                                              const float* __restrict__ X,
                                              float* BB,            // in: Bpre, out: B
                                              const float* __restrict__ bnet,
                                              float* __restrict__ A,
                                              float* __restrict__ NA,
                                              float* __restrict__ NB) {
  const int lane = threadIdx.x & 31;
  const int wave = threadIdx.x >> 5;
  const int row  = blockIdx.x * 8 + wave;
  float sa = 0.f, sb = 0.f;
#pragma unroll
  for (int q = 0; q < 4; ++q) {
    const int col = lane + q * 32;
    const int idx = row * MFEAT + col;
    const float x = X[idx];
    const float a = L[idx] * x;
    const float b = (BB[idx] + bnet[col]) * x;
    A[idx]  = a;
    BB[idx] = b;
    sa += a * a;
    sb += b * b;
  }
#pragma unroll
  for (int off = 16; off > 0; off >>= 1) {
    sa += __shfl_xor(sa, off, 32);
    sb += __shfl_xor(sb, off, 32);
  }
  if (lane == 0) { NA[row] = sa; NB[row] = sb; }
}

// --------------------------------------------------------------------------
// Kernel 3 (main, fused): for a 16-row i-tile, stream all j-tiles of a
// 1/JSPLIT slice: S = A_i . B_j^T via 32 chained v_wmma_f32_16x16x4_f32,
// then d2 = na+nb-2S -> w = exp(-sqrt(max(d2,1e-12))), accumulate
// wsum_i += w, wy_i += w*Y_j entirely in registers. Butterfly-reduce the
// 16 lanes that share a row, write one deterministic partial per (row,split).
// C layout: c[v] @ lane l = tile element (M = v + 8*(l>>4), N = l&15).
// --------------------------------------------------------------------------
__global__ __launch_bounds__(128) void k_pairwise(
    const float* __restrict__ A, const float* __restrict__ B,
    const float* __restrict__ NA, const float* __restrict__ NB,
    const float* __restrict__ Y, float* __restrict__ Pws,
    float* __restrict__ Pwy) {
  const int lane  = threadIdx.x & 31;
  const int wave  = threadIdx.x >> 5;
  const int js    = blockIdx.x & (JSPLIT - 1);        // waves in a block share
  const int iTile = (blockIdx.x >> 3) * 4 + wave;     // the same j-stream
  const int iBase = iTile * 16;
  const int half  = lane >> 4;
  const int lm    = lane & 15;

  // Resident A block: 16 rows x 128 features = 64 VGPRs/lane.
  v2f areg[32];
#pragma unroll
  for (int ks = 0; ks < 32; ++ks)
    areg[ks] = *(const v2f*)(A + (iBase + lm) * MFEAT + ks * 4 + 2 * half);

  float na8[8];
#pragma unroll
  for (int v = 0; v < 8; ++v) na8[v] = NA[iBase + 8 * half + v];

  float wsum[8], wy[8];
#pragma unroll
  for (int v = 0; v < 8; ++v) { wsum[v] = 0.f; wy[v] = 0.f; }

  const int jt0 = js * JTILES_PER_SPLIT;
  for (int jt = 0; jt < JTILES_PER_SPLIT; ++jt) {
    const int jBase = (jt0 + jt) * 16;
    const float nb = NB[jBase + lm];
    const float yj = Y[jBase + lm];
    v8f c = {};
#pragma unroll
    for (int ks = 0; ks < 32; ++ks) {
      v2f b = *(const v2f*)(B + (jBase + lm) * MFEAT + ks * 4 + 2 * half);
      c = __builtin_amdgcn_wmma_f32_16x16x4_f32(false, areg[ks], false, b,
                                                (short)0, c, false, false);
    }
#pragma unroll
    for (int v = 0; v < 8; ++v) {
      float d2 = na8[v] + nb - 2.0f * c[v];
      d2 = fmaxf(d2, 1e-12f);
      const float w = __expf(-sqrtf(d2));
      wsum[v] += w;
      wy[v]   += w * yj;
    }
  }

#pragma unroll
  for (int v = 0; v < 8; ++v) {
    float s = wsum[v], t = wy[v];
#pragma unroll
    for (int off = 8; off > 0; off >>= 1) {
      s += __shfl_xor(s, off, 16);
      t += __shfl_xor(t, off, 16);
    }
    if (lm == 0) {
      const int row = iBase + v + 8 * half;
      Pws[row * JSPLIT + js] = s;
      Pwy[row * JSPLIT + js] = t;
    }
  }
}

// --------------------------------------------------------------------------
// Kernel 4: combine partials, pred = wy/ws, loss = sum (Y - pred)^2.
// Single block, fixed-order tree -> bitwise deterministic.
// --------------------------------------------------------------------------
__global__ __launch_bounds__(1024) void k_combine(const float* __restrict__ Pws,
                                                  const float* __restrict__ Pwy,
                                                  const float* __restrict__ Y,
                                                  float* __restrict__ out) {
  __shared__ float red[1024];
  float acc = 0.f;
  for (int i = threadIdx.x; i < NROWS; i += 1024) {
    float ws = 0.f, wy = 0.f;
#pragma unroll
    for (int s = 0; s < JSPLIT; ++s) {
      ws += Pws[i * JSPLIT + s];
      wy += Pwy[i * JSPLIT + s];
    }
    const float d = Y[i] - wy / ws;
    acc += d * d;
  }
  red[threadIdx.x] = acc;
  __syncthreads();
  for (int s = 512; s > 0; s >>= 1) {
    if (threadIdx.x < s) red[threadIdx.x] += red[threadIdx.x + s];
    __syncthreads();
  }
  if (threadIdx.x == 0) out[0] = red[0];
}

// --------------------------------------------------------------------------
extern "C" void kernel_launch(void* const* d_in, const int* in_sizes, int n_in,
                              void* d_out, int out_size, void* d_ws,
                              size_t ws_size, hipStream_t stream) {
  const float* L    = (const float*)d_in[0];
  const float* X    = (const float*)d_in[1];
  const float* Y    = (const float*)d_in[2];
  const float* Wnet = (const float*)d_in[3];
  const float* bnet = (const float*)d_in[4];
  float* out = (float*)d_out;

  float* ws = (float*)d_ws;
  float* Bmat = ws;                                   // 8192*128 (Bpre, then B)
  float* Amat = Bmat + NROWS * MFEAT;                 // 8192*128
  float* NA   = Amat + NROWS * MFEAT;                 // 8192
  float* NB   = NA + NROWS;                           // 8192
  float* Pws  = NB + NROWS;                           // 8192*8
  float* Pwy  = Pws + NROWS * JSPLIT;                 // 8192*8

  // 1) Bpre = X @ Wnet  (4096 tiles, 4 waves/block)
  k_gemm_xw<<<1024, 128, 0, stream>>>(X, Wnet, Bmat);
  // 2) A, B (in place), row norms
  k_prep<<<NROWS / 8, 256, 0, stream>>>(L, X, Bmat, bnet, Amat, NA, NB);
  // 3) fused pairwise distance / kernel / NW accumulation
  k_pairwise<<<1024, 128, 0, stream>>>(Amat, Bmat, NA, NB, Y, Pws, Pwy);
  // 4) final deterministic reduction to scalar loss
  k_combine<<<1, 1024, 0, stream>>>(Pws, Pwy, Y, out);
}